// MultiGaussSpatialConv_66322884985175
// MI455X (gfx1250) — compile-verified
//
#include <hip/hip_runtime.h>

// MultiGaussSpatialConv fused kernel for MI455X (gfx1250, wave32).
//
// out[n,c] = sum_s w_s * (sum_m k_s[n,m]*yfea[m,c]) / (sum_m k_s[n,m])
//   k_s[n,m] = exp(-||x_n - y_m||^2 / (2 sigma_s^2))
//
// Each wave owns one 16-row n-tile and a contiguous M-segment. Inner loop over
// chunks of 4 m-values feeds V_WMMA_F32_16X16X4_F32:
//   A (16n x 4m) : lane = n-row, half-wave selects m-pair -> each lane computes
//                  its own two k = exp2(d2 * c2s) values (v_exp_f32).
//   B (4m x 16c) : lane = c, coalesced 64B yfea row loads.
//   D (16n x 16c): f32 accumulator carried across the whole segment.
// Denominators accumulate in VALU. Partials land in d_ws; a finalize kernel
// does a fixed-order (deterministic) reduction over segments + normalization.

typedef __attribute__((ext_vector_type(2))) float v2f;
typedef __attribute__((ext_vector_type(8))) float v8f;

#define WAVES_PER_BLOCK 8

__global__ __launch_bounds__(256) void mgsc_main(
    const float* __restrict__ x, const float* __restrict__ y,
    const float* __restrict__ yfea,
    float* __restrict__ wsNum,   // [nSeg][nTiles][3][16][16]
    float* __restrict__ wsDen,   // [nSeg][nTiles][3][2][16]
    float* __restrict__ outDirect, // used only when nSeg==1 (wsNum==nullptr)
    int nTiles, int nSeg, int mPerSeg)
{
    const int wave = blockIdx.x * WAVES_PER_BLOCK + (threadIdx.x >> 5);
    const int lane = threadIdx.x & 31;
    const int tile = wave / nSeg;
    const int seg  = wave % nSeg;
    if (tile >= nTiles) return;          // whole-wave uniform exit

    const int laneN = lane & 15;         // n-row within tile / c-column
    const int hi    = lane >> 4;         // half-wave: selects K={0,1} vs {2,3}

    // x row for this lane's n (resident for the whole kernel)
    const int n = tile * 16 + laneN;
    const float x0 = x[n * 3 + 0];
    const float x1 = x[n * 3 + 1];
    const float x2 = x[n * 3 + 2];

    // k = exp2(d2 * c2s), c2s = -log2(e)/(2 sigma^2)
    const float LOG2E = 1.4426950408889634f;
    const float c2s0 = -LOG2E / (2.0f * 0.05f * 0.05f);
    const float c2s1 = -LOG2E / (2.0f * 0.10f * 0.10f);
    const float c2s2 = -LOG2E / (2.0f * 0.20f * 0.20f);

    v8f acc0 = {}, acc1 = {}, acc2 = {};
    float den0 = 0.0f, den1 = 0.0f, den2 = 0.0f;

    const int mBeg = seg * mPerSeg;
    const int mEnd = mBeg + mPerSeg;

    #pragma unroll 2
    for (int m0 = mBeg; m0 < mEnd; m0 += 4) {
        const int ma = m0 + 2 * hi;      // this lane's two m values
        const int mb = ma + 1;

        // y rows (16 lanes share an address -> broadcast from L0/L2)
        const float ya0 = y[ma * 3 + 0], ya1 = y[ma * 3 + 1], ya2 = y[ma * 3 + 2];
        const float yb0 = y[mb * 3 + 0], yb1 = y[mb * 3 + 1], yb2 = y[mb * 3 + 2];

        // d2 via direct difference form (always >= 0, matches max(.,0) clamp)
        const float da0 = x0 - ya0, da1 = x1 - ya1, da2 = x2 - ya2;
        const float db0 = x0 - yb0, db1 = x1 - yb1, db2 = x2 - yb2;
        const float d2a = da0 * da0 + da1 * da1 + da2 * da2;
        const float d2b = db0 * db0 + db1 * db1 + db2 * db2;

        // B operand: yfea rows ma/mb, column = laneN (coalesced 64B per row)
        v2f B;
        B.x = yfea[ma * 16 + laneN];
        B.y = yfea[mb * 16 + laneN];

        // sigma 0
        {
            v2f A;
            A.x = __builtin_amdgcn_exp2f(d2a * c2s0);
            A.y = __builtin_amdgcn_exp2f(d2b * c2s0);
            den0 += A.x + A.y;
            acc0 = __builtin_amdgcn_wmma_f32_16x16x4_f32(
                false, A, false, B, (short)0, acc0, false, false);
        }
        // sigma 1
        {
            v2f A;
            A.x = __builtin_amdgcn_exp2f(d2a * c2s1);
            A.y = __builtin_amdgcn_exp2f(d2b * c2s1);
            den1 += A.x + A.y;
            acc1 = __builtin_amdgcn_wmma_f32_16x16x4_f32(
                false, A, false, B, (short)0, acc1, false, false);
        }
        // sigma 2
        {
            v2f A;
            A.x = __builtin_amdgcn_exp2f(d2a * c2s2);
            A.y = __builtin_amdgcn_exp2f(d2b * c2s2);
            den2 += A.x + A.y;
            acc2 = __builtin_amdgcn_wmma_f32_16x16x4_f32(
                false, A, false, B, (short)0, acc2, false, false);
        }
    }

    if (wsNum != nullptr) {
        // Partial path: dump D tiles in native layout + den partials to ws.
        const size_t baseT = ((size_t)(seg * nTiles + tile)) * 3;
        #pragma unroll
        for (int r = 0; r < 8; ++r) {
            const int row = r + 8 * hi;          // D layout: VGPR r, half-wave
            const size_t off = (size_t)row * 16 + laneN;
            wsNum[(baseT + 0) * 256 + off] = acc0[r];
            wsNum[(baseT + 1) * 256 + off] = acc1[r];
            wsNum[(baseT + 2) * 256 + off] = acc2[r];
        }
        const size_t doff = (size_t)hi * 16 + laneN;
        wsDen[(baseT + 0) * 32 + doff] = den0;
        wsDen[(baseT + 1) * 32 + doff] = den1;
        wsDen[(baseT + 2) * 32 + doff] = den2;
    } else {
        // Direct path (nSeg==1): combine den halves, broadcast per D row.
        den0 += __shfl_xor(den0, 16, 32);
        den1 += __shfl_xor(den1, 16, 32);
        den2 += __shfl_xor(den2, 16, 32);
        // now lane l holds den for n-row (l & 15)
        #pragma unroll
        for (int r = 0; r < 8; ++r) {
            const int row = r + 8 * hi;
            const float d0 = __shfl(den0, row, 32);
            const float d1 = __shfl(den1, row, 32);
            const float d2 = __shfl(den2, row, 32);
            const float o = 0.3f * acc0[r] / d0
                          + 0.3f * acc1[r] / d1
                          + 0.4f * acc2[r] / d2;
            outDirect[(size_t)(tile * 16 + row) * 16 + laneN] = o;
        }
    }
}

// Deterministic reduction over segments + normalization.
__global__ __launch_bounds__(256) void mgsc_final(
    const float* __restrict__ wsNum, const float* __restrict__ wsDen,
    float* __restrict__ out, int nTiles, int nSeg)
{
    const int idx = blockIdx.x * blockDim.x + threadIdx.x;   // n*16 + c
    const int total = nTiles * 256;
    if (idx >= total) return;
    const int c    = idx & 15;
    const int nrow = idx >> 4;
    const int tile = nrow >> 4;
    const int row  = nrow & 15;

    const float w[3] = {0.3f, 0.3f, 0.4f};
    float o = 0.0f;
    #pragma unroll
    for (int s = 0; s < 3; ++s) {
        float num = 0.0f, den = 0.0f;
        for (int seg = 0; seg < nSeg; ++seg) {
            const size_t b = ((size_t)(seg * nTiles + tile)) * 3 + s;
            num += wsNum[b * 256 + (size_t)row * 16 + c];
            den += wsDen[b * 32 + row] + wsDen[b * 32 + 16 + row];
        }
        o += w[s] * num / den;
    }
    out[idx] = o;
}

extern "C" void kernel_launch(void* const* d_in, const int* in_sizes, int n_in,
                              void* d_out, int out_size, void* d_ws, size_t ws_size,
                              hipStream_t stream) {
    const float* x  = (const float*)d_in[0];   // [1,N,3]
    const float* y  = (const float*)d_in[1];   // [1,M,3]
    const float* yf = (const float*)d_in[2];   // [1,M,16]
    float* out = (float*)d_out;

    const int N = in_sizes[0] / 3;
    const int M = in_sizes[1] / 3;
    const int nTiles = N / 16;

    // Pick the largest M-split whose partials fit in d_ws.
    // Per segment: num = nTiles*3*256 floats, den = nTiles*3*32 floats.
    const size_t numPerSeg = (size_t)nTiles * 3 * 256;
    const size_t denPerSeg = (size_t)nTiles * 3 * 32;
    const size_t bytesPerSeg = (numPerSeg + denPerSeg) * sizeof(float);
    int nSeg = 1;
    const int cands[4] = {16, 8, 4, 2};
    for (int i = 0; i < 4; ++i) {
        const int c = cands[i];
        if ((size_t)c * bytesPerSeg <= ws_size && (M % (c * 4)) == 0) {
            nSeg = c;
            break;
        }
    }
    const int mPerSeg = M / nSeg;

    float* wsNum = nullptr;
    float* wsDen = nullptr;
    if (nSeg > 1) {
        wsNum = (float*)d_ws;
        wsDen = wsNum + (size_t)nSeg * numPerSeg;
    }

    const int totalWaves = nTiles * nSeg;
    const int blocks = (totalWaves + WAVES_PER_BLOCK - 1) / WAVES_PER_BLOCK;
    mgsc_main<<<blocks, 256, 0, stream>>>(x, y, yf, wsNum, wsDen,
                                          (nSeg > 1) ? nullptr : out,
                                          nTiles, nSeg, mPerSeg);
    if (nSeg > 1) {
        const int totalOut = nTiles * 256;
        const int fblocks = (totalOut + 255) / 256;
        mgsc_final<<<fblocks, 256, 0, stream>>>(wsNum, wsDen, out, nTiles, nSeg);
    }
}